// GNN_18743237279940
// MI455X (gfx1250) — compile-verified
//
#include <hip/hip_runtime.h>
#include <math.h>

typedef __attribute__((ext_vector_type(2))) float v2f;
typedef __attribute__((ext_vector_type(8))) float v8f;

// Non-returning HW float atomic (global_atomic_add_f32, agent scope, relaxed).
__device__ __forceinline__ void atomAddF(float* p, float v) {
    __hip_atomic_fetch_add(p, v, __ATOMIC_RELAXED, __HIP_MEMORY_SCOPE_AGENT);
}

// ---------------- Pass 1: degrees ----------------
__global__ __launch_bounds__(256) void k_deg_init(int* __restrict__ deg, int n) {
    int i = blockIdx.x * blockDim.x + threadIdx.x;
    if (i < n) deg[i] = 1;  // self-loop
}

__global__ __launch_bounds__(256) void k_edge_deg(const int* __restrict__ dst,
                                                  int* __restrict__ deg, int E) {
    int i = blockIdx.x * blockDim.x + threadIdx.x;
    int e = i * 4;
    if (e + 4 <= E) {
        int4 c = *reinterpret_cast<const int4*>(dst + e);
        atomicAdd(&deg[c.x], 1);
        atomicAdd(&deg[c.y], 1);
        atomicAdd(&deg[c.z], 1);
        atomicAdd(&deg[c.w], 1);
    } else {
        for (; e < E; ++e) atomicAdd(&deg[dst[e]], 1);
    }
}

// ---------------- Pass 2: dis + layer-1 self-loop init ----------------
__global__ __launch_bounds__(256) void k_node_prep(const int* __restrict__ deg,
                                                   const float* __restrict__ x,
                                                   float* __restrict__ dis,
                                                   float* __restrict__ s, int n) {
    int i = blockIdx.x * blockDim.x + threadIdx.x;
    if (i < n) {
        float r = rsqrtf((float)deg[i]);
        dis[i] = r;
        s[i] = x[i] * (r * r);  // self-loop norm = dis^2
    }
}

// ---------------- Pass 3: layer-1 edge aggregation (scalar) ----------------
__global__ __launch_bounds__(256) void k_edge_agg1(const int* __restrict__ src,
                                                   const int* __restrict__ dst,
                                                   const float* __restrict__ x,
                                                   const float* __restrict__ dis,
                                                   float* __restrict__ s, int E) {
    int i = blockIdx.x * blockDim.x + threadIdx.x;
    int e = i * 4;
    if (e + 4 <= E) {
        int4 r = *reinterpret_cast<const int4*>(src + e);
        int4 c = *reinterpret_cast<const int4*>(dst + e);
        atomAddF(&s[c.x], x[r.x] * dis[r.x] * dis[c.x]);
        atomAddF(&s[c.y], x[r.y] * dis[r.y] * dis[c.y]);
        atomAddF(&s[c.z], x[r.z] * dis[r.z] * dis[c.z]);
        atomAddF(&s[c.w], x[r.w] * dis[r.w] * dis[c.w]);
    } else {
        for (; e < E; ++e) {
            int r = src[e], c = dst[e];
            atomAddF(&s[c], x[r] * dis[r] * dis[c]);
        }
    }
}

// ---------------- Pass 4: dense transform z = relu(s*W1+b1) @ W2 via WMMA ----
// One wave handles 16 nodes. H tile is 16x16 (M=node, K=hidden), W2 padded to
// 16x16 (N=class, cols 2..15 zero). K=16 done as 4 chained V_WMMA_F32_16X16X4_F32.
// All operand prep is hoisted before the WMMA chain; W2 padding is a value
// select (v_cndmask) on an unconditional clamped load — no exec-mask branches.
__global__ __launch_bounds__(256) void k_node_z(const float* __restrict__ s,
                                                const float* __restrict__ dis,
                                                const float* __restrict__ W1,
                                                const float* __restrict__ b1,
                                                const float* __restrict__ W2,
                                                float2* __restrict__ z,
                                                float2* __restrict__ acc, int n) {
    __shared__ float tile[8][256];  // 8 waves/block, 16x16 f32 each
    const int wave = threadIdx.x >> 5;
    const int lane = threadIdx.x & 31;
    const int m    = lane & 15;   // row within tile (A) / N index (B)
    const int half = lane >> 4;   // K-group selector
    const int base = (blockIdx.x * 8 + wave) * 16;
    if (base >= n) return;        // wave-uniform branch: no EXEC masking at WMMA

    int node = base + m;
    if (node >= n) node = n - 1;  // clamp (dup rows are discarded on write)
    const float sv = s[node];

    // Hoisted operand prep for all 4 K-blocks.
    float av0[4], av1[4], bv0[4], bv1[4];
    const int mc = (m < 2) ? m : 1;          // clamped, always in-bounds
    const float pad = (m < 2) ? 1.0f : 0.0f; // zero-pad cols 2..15 of W2
#pragma unroll
    for (int kb = 0; kb < 4; ++kb) {
        // A 16x4 layout: lanes 0-15 hold K={0,1}, lanes 16-31 hold K={2,3}
        const int j0 = kb * 4 + half * 2;
        const int j1 = j0 + 1;
        av0[kb] = fmaxf(fmaf(sv, W1[j0], b1[j0]), 0.0f);
        av1[kb] = fmaxf(fmaf(sv, W1[j1], b1[j1]), 0.0f);
        // B 4x16 layout mirrors A: per-lane N = m, K groups split at lane 16
        bv0[kb] = W2[j0 * 2 + mc] * pad;
        bv1[kb] = W2[j1 * 2 + mc] * pad;
    }

    v8f c = {};  // 16x16 f32 accumulator
#pragma unroll
    for (int kb = 0; kb < 4; ++kb) {
        v2f a; a.x = av0[kb]; a.y = av1[kb];
        v2f b; b.x = bv0[kb]; b.y = bv1[kb];
        c = __builtin_amdgcn_wmma_f32_16x16x4_f32(false, a, false, b,
                                                  (short)0, c, false, false);
    }

    // D layout: c[i] holds element (M = i + 8*half, N = m). Bounce via LDS.
    float* t = tile[wave];
#pragma unroll
    for (int i = 0; i < 8; ++i) t[(i + 8 * half) * 16 + m] = c[i];
    asm volatile("s_wait_dscnt 0" ::: "memory");

    if (half == 0) {
        int nd = base + m;
        if (nd < n) {
            float z0 = t[m * 16 + 0];
            float z1 = t[m * 16 + 1];
            float dv = dis[nd];
            float d2 = dv * dv;
            z[nd]   = make_float2(z0, z1);
            acc[nd] = make_float2(z0 * d2, z1 * d2);  // layer-2 self-loop init
        }
    }
}

// ---------------- Pass 5: layer-2 edge aggregation (2-vector) ----------------
__global__ __launch_bounds__(256) void k_edge_agg2(const int* __restrict__ src,
                                                   const int* __restrict__ dst,
                                                   const float2* __restrict__ z,
                                                   const float* __restrict__ dis,
                                                   float2* __restrict__ acc, int E) {
    int i = blockIdx.x * blockDim.x + threadIdx.x;
    int e = i * 4;
    if (e + 4 <= E) {
        int4 r = *reinterpret_cast<const int4*>(src + e);
        int4 c = *reinterpret_cast<const int4*>(dst + e);
        {
            float2 zr = z[r.x]; float nm = dis[r.x] * dis[c.x];
            atomAddF(&acc[c.x].x, zr.x * nm); atomAddF(&acc[c.x].y, zr.y * nm);
        }
        {
            float2 zr = z[r.y]; float nm = dis[r.y] * dis[c.y];
            atomAddF(&acc[c.y].x, zr.x * nm); atomAddF(&acc[c.y].y, zr.y * nm);
        }
        {
            float2 zr = z[r.z]; float nm = dis[r.z] * dis[c.z];
            atomAddF(&acc[c.z].x, zr.x * nm); atomAddF(&acc[c.z].y, zr.y * nm);
        }
        {
            float2 zr = z[r.w]; float nm = dis[r.w] * dis[c.w];
            atomAddF(&acc[c.w].x, zr.x * nm); atomAddF(&acc[c.w].y, zr.y * nm);
        }
    } else {
        for (; e < E; ++e) {
            int r = src[e], c = dst[e];
            float2 zr = z[r]; float nm = dis[r] * dis[c];
            atomAddF(&acc[c].x, zr.x * nm); atomAddF(&acc[c].y, zr.y * nm);
        }
    }
}

// ---------------- Pass 6: bias + 2-class log-softmax ----------------
__global__ __launch_bounds__(256) void k_final(const float2* __restrict__ acc,
                                               const float* __restrict__ b2,
                                               float* __restrict__ out, int n) {
    int i = blockIdx.x * blockDim.x + threadIdx.x;
    if (i < n) {
        float v0 = acc[i].x + b2[0];
        float v1 = acc[i].y + b2[1];
        float mx = fmaxf(v0, v1);
        float lse = mx + logf(expf(v0 - mx) + expf(v1 - mx));
        out[2 * i + 0] = v0 - lse;
        out[2 * i + 1] = v1 - lse;
    }
}

extern "C" void kernel_launch(void* const* d_in, const int* in_sizes, int n_in,
                              void* d_out, int out_size, void* d_ws, size_t ws_size,
                              hipStream_t stream) {
    (void)n_in; (void)out_size; (void)ws_size;
    const float* x  = (const float*)d_in[0];
    const int*   ei = (const int*)d_in[1];
    const float* W1 = (const float*)d_in[2];
    const float* b1 = (const float*)d_in[3];
    const float* W2 = (const float*)d_in[4];
    const float* b2 = (const float*)d_in[5];
    const int N = in_sizes[0];       // x is [N,1]
    const int E = in_sizes[1] / 2;   // edge_index is [2,E]
    const int* src = ei;             // edge_index[0] (source / row)
    const int* dst = ei + E;         // edge_index[1] (target / col)

    // Workspace layout (256B-aligned slices): deg | dis | s | z | acc
    char* ws = (char*)d_ws;
    auto align = [](size_t v) { return (v + 255) & ~size_t(255); };
    size_t o = 0;
    int*    deg = (int*)(ws + o);    o = align(o + sizeof(int)    * (size_t)N);
    float*  dis = (float*)(ws + o);  o = align(o + sizeof(float)  * (size_t)N);
    float*  s   = (float*)(ws + o);  o = align(o + sizeof(float)  * (size_t)N);
    float2* z   = (float2*)(ws + o); o = align(o + sizeof(float2) * (size_t)N);
    float2* acc = (float2*)(ws + o);

    const int TB = 256;
    const int nodeBlocks = (N + TB - 1) / TB;
    const int edgeThreads = (E + 3) / 4;
    const int edgeBlocks = (edgeThreads + TB - 1) / TB;
    const int tileBlocks = (((N + 15) / 16) + 7) / 8;  // 16 nodes/wave, 8 waves/block

    k_deg_init <<<nodeBlocks, TB, 0, stream>>>(deg, N);
    k_edge_deg <<<edgeBlocks, TB, 0, stream>>>(dst, deg, E);
    k_node_prep<<<nodeBlocks, TB, 0, stream>>>(deg, x, dis, s, N);
    k_edge_agg1<<<edgeBlocks, TB, 0, stream>>>(src, dst, x, dis, s, E);
    k_node_z   <<<tileBlocks, TB, 0, stream>>>(s, dis, W1, b1, W2, z, acc, N);
    k_edge_agg2<<<edgeBlocks, TB, 0, stream>>>(src, dst, z, dis, acc, E);
    k_final    <<<nodeBlocks, TB, 0, stream>>>(acc, b2, (float*)d_out, N);
}